// FeatureExtractionStyleTransferModel_50620484551217
// MI455X (gfx1250) — compile-verified
//
#include <hip/hip_runtime.h>
#include <hip/hip_bf16.h>

typedef __attribute__((ext_vector_type(2))) float v2f;
typedef __attribute__((ext_vector_type(8))) float v8f;
typedef unsigned __attribute__((ext_vector_type(4))) u32x4;
typedef int __attribute__((ext_vector_type(4))) i32x4;
typedef int __attribute__((ext_vector_type(8))) i32x8;

#define N_BATCH 4
#define C_OUT   64
#define HH      512
#define WW      512
#define S_LEN   (HH * WW)            /* 262144 per channel       */
#define M_LEN   (448 * 448)          /* 200704 per ref channel   */
#define NC      (N_BATCH * C_OUT)    /* 256 channels             */
#define FEAT_ELEMS ((size_t)N_BATCH * C_OUT * S_LEN) /* 67108864 */

// ---------- float <-> radix-sortable key ----------
__device__ __forceinline__ unsigned keyFromFloat(float f) {
  unsigned u = __float_as_uint(f);
  return u ^ ((u >> 31) ? 0xFFFFFFFFu : 0x80000000u);
}
__device__ __forceinline__ float floatFromKey(unsigned k) {
  unsigned u = (k >> 31) ? (k ^ 0x80000000u) : (k ^ 0xFFFFFFFFu);
  return __uint_as_float(u);
}

// ---------- TDM: stage one 64x64 fp32 tile (row stride S_LEN) into LDS ----------
// D# group0: count=1 | lds_addr | 57b global_addr | type=2 ("image")
// D# group1: data_size=4B, pad_enable, pad_interval=64 DWORDs (code 5),
//            pad_amount=1 DWORD (code 0)  ->  LDS row stride 65 DWORDs,
//            tensor_dim0=S_LEN, tensor_dim1=64, tile_dim0=64, tile_dim1=64,
//            tensor_dim0_stride=S_LEN.
__device__ __forceinline__ void tdm_stage_tile(const float* gsrc, unsigned lds_addr) {
  unsigned long long ga = (unsigned long long)(size_t)gsrc;
  u32x4 g0;
  g0.x = 1u;                                             // count=1, user descriptor
  g0.y = lds_addr;                                       // LDS byte address
  g0.z = (unsigned)(ga & 0xFFFFFFFFu);                   // global_addr[31:0]
  g0.w = (unsigned)((ga >> 32) & 0x01FFFFFFu)            // global_addr[56:32]
         | 0x80000000u;                                  // type=2
  i32x8 g1;
  g1[0] = (2 << 16) | (1 << 20) | (5 << 22);             // dsize=4B, pad on, 64dw/+1dw
  g1[1] = 0;                                             // abar=0, tensor_dim0[15:0]=0
  g1[2] = (int)(((unsigned)S_LEN >> 16) | (64u << 16));  // tdim0 hi | tdim1 lo=64
  g1[3] = (int)(64u << 16);                              // tdim1 hi=0 | tile_dim0=64
  g1[4] = 64;                                            // tile_dim1=64, tile_dim2=0
  g1[5] = (int)S_LEN;                                    // tensor_dim0_stride lo
  g1[6] = 0;                                             // stride hi / dim1_stride
  g1[7] = 0;
  i32x4 z4 = {};
#if __clang_major__ >= 23
  i32x8 z8 = {};
  __builtin_amdgcn_tensor_load_to_lds(g0, g1, z4, z4, z8, 0);
#else
  __builtin_amdgcn_tensor_load_to_lds(g0, g1, z4, z4, 0);
#endif
}

// ---------- zero accumulators + gram ----------
__global__ void init_ws(float* accum, float* gram) {
  int i = blockIdx.x * 256 + threadIdx.x;
  if (i < 8) accum[i] = 0.0f;
  if (i < N_BATCH * C_OUT * C_OUT) gram[i] = 0.0f;
}

// ---------- normalize + conv3x3 (3->64) + bias + relu, fused content loss ----------
__global__ __launch_bounds__(256) void conv_relu_content(
    const float* __restrict__ img, const float* __restrict__ cw,
    const float* __restrict__ cb, const float* __restrict__ ct,
    float* __restrict__ feat, float* __restrict__ accum) {
  const float mean[3]   = {0.485f, 0.456f, 0.406f};
  const float invstd[3] = {1.0f / 0.229f, 1.0f / 0.224f, 1.0f / 0.225f};
  size_t idx = (size_t)blockIdx.x * 256 + threadIdx.x;
  int x  = (int)(idx & (WW - 1));
  int y  = (int)((idx >> 9) & (HH - 1));
  int co = (int)((idx >> 18) & (C_OUT - 1));
  int n  = (int)(idx >> 24);

  float sum = cb[co];
#pragma unroll
  for (int ci = 0; ci < 3; ++ci) {
#pragma unroll
    for (int ky = 0; ky < 3; ++ky) {
      int yy = y + ky - 1;
#pragma unroll
      for (int kx = 0; kx < 3; ++kx) {
        int xx = x + kx - 1;
        float v = 0.0f;
        if (yy >= 0 && yy < HH && xx >= 0 && xx < WW) {
          float p = img[(((size_t)n * 3 + ci) * HH + yy) * WW + xx];
          v = (p - mean[ci]) * invstd[ci];  // SAME padding pads *normalized* input with 0
        }
        sum += cw[((co * 3 + ci) * 3 + ky) * 3 + kx] * v;
      }
    }
  }
  float f = fmaxf(sum, 0.0f);
  feat[idx] = f;

  float d = f - ct[idx];
  __shared__ float red[256];
  red[threadIdx.x] = d * d;
  __syncthreads();
  for (int off = 128; off > 0; off >>= 1) {
    if (threadIdx.x < off) red[threadIdx.x] += red[threadIdx.x + off];
    __syncthreads();
  }
  if (threadIdx.x == 0) atomicAdd(accum + 0, red[0]);
}

// ---------- Gram = F * F^T per batch via fp32 WMMA 16x16x4, TDM-staged LDS ----------
// grid: (64 s-chunks, N). Wave 0 DMAs a 64x64 fp32 slab of F into LDS via the
// Tensor Data Mover (with hardware 65-DWORD-row padding), 8 waves each own two
// 16x16 output tiles, K advances 4 per v_wmma.
__global__ __launch_bounds__(256) void gram_wmma(
    const float* __restrict__ feat, float* __restrict__ gram) {
  __shared__ float tile[64][65];  // row stride 65 -> conflict-free column reads
  const int n     = blockIdx.y;
  const int chunk = blockIdx.x;          // 0..63, each covers 4096 columns of S
  const int t     = threadIdx.x;
  const int wave  = t >> 5;
  const int lane  = t & 31;
  const int lm    = lane & 15;           // M (for A) / N (for B) index
  const int hi    = lane >> 4;           // lanes 16-31 hold K+2/K+3 halves
  const int koff  = 2 * hi;
  const int mi0 = wave >> 2, ni0 = wave & 3;
  const int w1  = wave + 8;
  const int mi1 = w1 >> 2,  ni1 = w1 & 3;

  const float* F = feat + (size_t)n * C_OUT * S_LEN;
  const unsigned ldsTileAddr = (unsigned)(size_t)&tile[0][0];  // LDS offset = addr[31:0]
  v8f acc0 = {};
  v8f acc1 = {};

  for (int it = 0; it < 64; ++it) {
    int s0 = chunk * 4096 + it * 64;
    __syncthreads();  // all waves done reading previous tile
    if (t < 32) {     // single wave issues the tensor DMA and waits on it
      tdm_stage_tile(F + s0, ldsTileAddr);
      __builtin_amdgcn_s_wait_tensorcnt(0);
    }
    __syncthreads();  // tile visible to all waves
#pragma unroll 4
    for (int kk = 0; kk < 64; kk += 4) {
      v2f a0, b0, a1, b1;
      a0.x = tile[mi0 * 16 + lm][kk + koff];
      a0.y = tile[mi0 * 16 + lm][kk + koff + 1];
      b0.x = tile[ni0 * 16 + lm][kk + koff];
      b0.y = tile[ni0 * 16 + lm][kk + koff + 1];
      a1.x = tile[mi1 * 16 + lm][kk + koff];
      a1.y = tile[mi1 * 16 + lm][kk + koff + 1];
      b1.x = tile[ni1 * 16 + lm][kk + koff];
      b1.y = tile[ni1 * 16 + lm][kk + koff + 1];
      acc0 = __builtin_amdgcn_wmma_f32_16x16x4_f32(false, a0, false, b0,
                                                   (short)0, acc0, false, false);
      acc1 = __builtin_amdgcn_wmma_f32_16x16x4_f32(false, a1, false, b1,
                                                   (short)0, acc1, false, false);
    }
  }

  float* G = gram + (size_t)n * C_OUT * C_OUT;
#pragma unroll
  for (int v = 0; v < 8; ++v) {
    int r0 = mi0 * 16 + v + hi * 8, c0 = ni0 * 16 + lm;
    atomicAdd(&G[r0 * C_OUT + c0], acc0[v]);
    int r1 = mi1 * 16 + v + hi * 8, c1 = ni1 * 16 + lm;
    atomicAdd(&G[r1 * C_OUT + c1], acc1[v]);
  }
}

// ---------- style loss ----------
__global__ void style_loss(const float* __restrict__ gram,
                           const float* __restrict__ tgt, float* accum) {
  int idx = blockIdx.x * 256 + threadIdx.x;  // 0..16383
  float g = gram[idx] * (1.0f / ((float)C_OUT * (float)S_LEN));
  float d = g - tgt[idx];
  __shared__ float red[256];
  red[threadIdx.x] = d * d;
  __syncthreads();
  for (int off = 128; off > 0; off >>= 1) {
    if (threadIdx.x < off) red[threadIdx.x] += red[threadIdx.x + off];
    __syncthreads();
  }
  if (threadIdx.x == 0) atomicAdd(accum + 1, red[0]);
}

// ---------- per-channel keys-only LSD radix sort (one workgroup per channel) ----------
template <int L>
__global__ __launch_bounds__(256) void radix_sort_channel(
    const float* __restrict__ vals, unsigned* __restrict__ bufA,
    unsigned* __restrict__ bufB) {
  constexpr int T = 256;
  constexpr int CHUNK = L / T;  // 1024 (src) / 784 (ref), both exact
  const int ch = blockIdx.x;
  const int t  = threadIdx.x;
  const float* src = vals + (size_t)ch * L;
  unsigned* A = bufA + (size_t)ch * L;
  unsigned* B = bufB + (size_t)ch * L;

  __shared__ unsigned hist[256][17];  // [thread][digit], padded
  __shared__ unsigned base[16];

  // transform floats into monotonic uint keys
  for (int i = 0; i < CHUNK; ++i) {
    int e = t * CHUNK + i;
    A[e] = keyFromFloat(src[e]);
  }
  __threadfence_block();
  __syncthreads();

  for (int p = 0; p < 8; ++p) {  // 8 passes x 4 bits; result ends in A
    unsigned* in  = (p & 1) ? B : A;
    unsigned* out = (p & 1) ? A : B;
    const int shift = p * 4;

    for (int d = 0; d < 16; ++d) hist[t][d] = 0;
    __syncthreads();
    for (int i = 0; i < CHUNK; ++i) {
      unsigned k = in[t * CHUNK + i];
      if ((i & 63) == 0) __builtin_prefetch(&in[t * CHUNK + i + 64], 0, 1);
      hist[t][(k >> shift) & 15]++;
    }
    __syncthreads();
    if (t < 16) {  // per-digit exclusive prefix over threads (stability)
      unsigned run = 0;
      for (int tt = 0; tt < 256; ++tt) {
        unsigned c = hist[tt][t];
        hist[tt][t] = run;
        run += c;
      }
      base[t] = run;
    }
    __syncthreads();
    if (t == 0) {  // exclusive scan of digit totals
      unsigned run = 0;
      for (int d = 0; d < 16; ++d) { unsigned c = base[d]; base[d] = run; run += c; }
    }
    __syncthreads();
    for (int i = 0; i < CHUNK; ++i) {  // stable scatter
      unsigned k = in[t * CHUNK + i];
      int d = (k >> shift) & 15;
      unsigned pos = base[d] + hist[t][d]++;
      out[pos] = k;
    }
    __threadfence_block();
    __syncthreads();
  }
}

// ---------- histogram-matching loss: loss = mean_r (sorted_src[r] - interp(sorted_ref))^2 ----------
__global__ void hist_loss(const unsigned* __restrict__ srcSorted,
                          const unsigned* __restrict__ refSorted, float* accum) {
  size_t idx = (size_t)blockIdx.x * 256 + threadIdx.x;
  int r  = (int)(idx & (S_LEN - 1));
  int ch = (int)(idx >> 18);
  const float scale = (float)(M_LEN - 1) / (float)(S_LEN - 1);
  float pos = (float)r * scale;
  int lo = (int)pos;
  if (lo > M_LEN - 1) lo = M_LEN - 1;
  int hi = lo + 1;
  if (hi > M_LEN - 1) hi = M_LEN - 1;
  float w = pos - (float)lo;

  const unsigned* ss = srcSorted + (size_t)ch * S_LEN;
  const unsigned* rs = refSorted + (size_t)ch * M_LEN;
  float sv  = floatFromKey(ss[r]);
  float rl  = floatFromKey(rs[lo]);
  float rh  = floatFromKey(rs[hi]);
  float res = rl * (1.0f - w) + rh * w;
  float d   = sv - res;

  __shared__ float red[256];
  red[threadIdx.x] = d * d;
  __syncthreads();
  for (int off = 128; off > 0; off >>= 1) {
    if (threadIdx.x < off) red[threadIdx.x] += red[threadIdx.x + off];
    __syncthreads();
  }
  if (threadIdx.x == 0) atomicAdd(accum + 2, red[0]);
}

// ---------- finalize scalar losses ----------
__global__ void finalize(const float* __restrict__ accum, float* __restrict__ out) {
  if (blockIdx.x == 0 && threadIdx.x == 0) {
    out[0] = accum[0] / (float)FEAT_ELEMS;                       // content
    out[1] = accum[1] / (float)(N_BATCH * C_OUT * C_OUT);        // style
    out[2] = accum[2] / (float)FEAT_ELEMS;                       // hist
  }
}

extern "C" void kernel_launch(void* const* d_in, const int* in_sizes, int n_in,
                              void* d_out, int out_size, void* d_ws, size_t ws_size,
                              hipStream_t stream) {
  const float* img = (const float*)d_in[0];
  const float* cw  = (const float*)d_in[1];
  const float* cb  = (const float*)d_in[2];
  const float* ct  = (const float*)d_in[3];
  const float* sg  = (const float*)d_in[4];
  const float* ht  = (const float*)d_in[5];

  float* feat   = (float*)d_out;
  float* losses = feat + FEAT_ELEMS;

  // workspace layout
  float*    accum = (float*)d_ws;                          // 8 floats
  float*    gram  = accum + 8;                             // 4*64*64
  unsigned* srcA  = (unsigned*)(gram + N_BATCH * C_OUT * C_OUT);
  unsigned* srcB  = srcA + FEAT_ELEMS;
  unsigned* refA  = srcB + FEAT_ELEMS;
  unsigned* refB  = refA + (size_t)NC * M_LEN;

  init_ws<<<64, 256, 0, stream>>>(accum, gram);
  conv_relu_content<<<(unsigned)(FEAT_ELEMS / 256), 256, 0, stream>>>(
      img, cw, cb, ct, feat, accum);
  gram_wmma<<<dim3(64, N_BATCH), 256, 0, stream>>>(feat, gram);
  style_loss<<<(N_BATCH * C_OUT * C_OUT) / 256, 256, 0, stream>>>(gram, sg, accum);
  radix_sort_channel<S_LEN><<<NC, 256, 0, stream>>>(feat, srcA, srcB);
  radix_sort_channel<M_LEN><<<NC, 256, 0, stream>>>(ht, refA, refB);
  hist_loss<<<(unsigned)(FEAT_ELEMS / 256), 256, 0, stream>>>(srcA, refA, accum);
  finalize<<<1, 1, 0, stream>>>(accum, losses);
}